// GAT_68109591380379
// MI455X (gfx1250) — compile-verified
//
#include <hip/hip_runtime.h>
#include <cstdint>
#include <math.h>

#define NEG_SLOPE 0.2f

typedef float v2f __attribute__((ext_vector_type(2)));
typedef float v8f __attribute__((ext_vector_type(8)));
typedef unsigned int v4u __attribute__((ext_vector_type(4)));
typedef int v4i __attribute__((ext_vector_type(4)));
typedef int v8i __attribute__((ext_vector_type(8)));

// ---------------- WMMA helper: D = A(16x4,f32) * B(4x16,f32) + C ----------------
__device__ __forceinline__ v8f wmma_f32_4(v2f a, v2f b, v8f c) {
#if defined(__HIP_DEVICE_COMPILE__) && __has_builtin(__builtin_amdgcn_wmma_f32_16x16x4_f32)
  // (neg_a, A, neg_b, B, c_mod, C, reuse_a, reuse_b)
  return __builtin_amdgcn_wmma_f32_16x16x4_f32(false, a, false, b, (short)0, c,
                                               false, false);
#else
  return c;
#endif
}

// ---------------- TDM: DMA a dense 2-D fp32 matrix (dim1 x dim0) into LDS ------
// dim0 = fastest (contiguous) dimension in elements, stride0 in elements.
__device__ __forceinline__ void lds_load_matrix(const float* __restrict__ g,
                                                float* sdst, int elems,
                                                uint32_t dim0, uint32_t dim1,
                                                uint32_t stride0) {
#if defined(__HIP_DEVICE_COMPILE__) && __has_builtin(__builtin_amdgcn_tensor_load_to_lds)
  (void)elems;
  if (threadIdx.x < 32) {  // one wave issues the DMA
    uint64_t ga = (uint64_t)(uintptr_t)g;
    uint32_t lo = (uint32_t)(uintptr_t)sdst;  // generic->LDS offset (low 32 bits)
    v4u g0;
    g0.x = 1u;                                    // count=1, user descriptor
    g0.y = lo;                                    // lds_addr
    g0.z = (uint32_t)ga;                          // global_addr[31:0]
    g0.w = ((uint32_t)(ga >> 32) & 0x01FFFFFFu)   // global_addr[56:32]
           | (2u << 30);                          // type=2 ("image")
    v8i g1;
    g1[0] = (int)(2u << 16);                      // data_size=2 (4 bytes)
    g1[1] = (int)((dim0 & 0xFFFFu) << 16);        // tensor_dim0[15:0]
    g1[2] = (int)((dim0 >> 16) | ((dim1 & 0xFFFFu) << 16));  // dim0 hi, dim1 lo
    g1[3] = (int)((dim1 >> 16) | ((dim0 & 0xFFFFu) << 16));  // dim1 hi, tile_dim0
    g1[4] = (int)(dim1 & 0xFFFFu);                // tile_dim1 (tile_dim2=0)
    g1[5] = (int)stride0;                         // tensor_dim0_stride low 32
    g1[6] = 0;
    g1[7] = 0;
    v4i z = {0, 0, 0, 0};
#if __clang_major__ >= 23
    v8i z8 = {0, 0, 0, 0, 0, 0, 0, 0};
    __builtin_amdgcn_tensor_load_to_lds(g0, g1, z, z, z8, 0);
#else
    __builtin_amdgcn_tensor_load_to_lds(g0, g1, z, z, 0);
#endif
    __builtin_amdgcn_s_wait_tensorcnt(0);
  }
#else
  for (int i = threadIdx.x; i < elems; i += blockDim.x) sdst[i] = g[i];
#endif
  __syncthreads();
}

// ---------------- misc device helpers ----------------
__device__ __forceinline__ float lrelu(float e) {
  return e > 0.f ? e : NEG_SLOPE * e;
}
__device__ __forceinline__ void atomicMaxF(float* a, float v) {
  // correct for any mix of signs when *a starts at -inf
  if (v >= 0.f)
    atomicMax((int*)a, __float_as_int(v));
  else
    atomicMin((unsigned int*)a, __float_as_uint(v));
}

// ---------------- init: seg-max = -inf, denom = 0, accumulators = 0 ------------
__global__ void gat_init(float* m1, float* d1, float* m2, float* d2,
                         float* agg1, float* agg2, int N) {
  int i = blockIdx.x * blockDim.x + threadIdx.x;
  if (i >= 32 * N) return;
  if (i < N) {
    m1[i] = -INFINITY;
    d1[i] = 0.f;
    m2[i] = -INFINITY;
    d2[i] = 0.f;
  }
  if (i < 16 * N) agg1[i] = 0.f;
  agg2[i] = 0.f;
}

// ---------------- GEMM1: h1[N,16] = x[N,512] @ W1[512,16] (WMMA f32) -----------
__global__ __launch_bounds__(256) void gat_gemm1(const float* __restrict__ x,
                                                 const float* __restrict__ W1,
                                                 float* __restrict__ h1, int N) {
  __shared__ float sW[512 * 16];  // 32 KB, DMA'd by TDM
  lds_load_matrix(W1, sW, 512 * 16, /*dim0=*/16, /*dim1=*/512, /*stride0=*/16);

  const int wave = threadIdx.x >> 5;
  const int lane = threadIdx.x & 31;
  const int half = lane >> 4;  // selects K pair {0,1} vs {2,3}
  const int n = lane & 15;     // M for A-loads, N for B/C
  const int rowbase = (blockIdx.x * 8 + wave) * 16;

  int rowa = rowbase + n;
  if (rowa >= N) rowa = N - 1;  // clamp loads; EXEC must stay all-1 for WMMA
  const float* xrow = x + (size_t)rowa * 512 + 2 * half;

  v8f c = {};
#pragma unroll 4
  for (int k0 = 0; k0 < 512; k0 += 4) {
    float2 t = *(const float2*)(xrow + k0);  // K = k0+2*half, +1
    v2f a;
    a.x = t.x;
    a.y = t.y;
    v2f b;
    b.x = sW[(k0 + 2 * half + 0) * 16 + n];
    b.y = sW[(k0 + 2 * half + 1) * 16 + n];
    c = wmma_f32_4(a, b, c);
  }
  float* o = h1 + (size_t)(rowbase + 8 * half) * 16 + n;
  if (rowbase + 16 <= N) {  // fast path: full tile, no guards (all but last block)
#pragma unroll
    for (int r = 0; r < 8; ++r) o[(size_t)r * 16] = c[r];
  } else {
#pragma unroll
    for (int r = 0; r < 8; ++r) {
      int rowo = rowbase + r + 8 * half;  // C layout: VGPR r -> M=r / M=r+8
      if (rowo < N) h1[(size_t)rowo * 16 + n] = c[r];
    }
  }
}

// ---------------- GEMM2: h2[N,32] = hrelu[N,16] @ W2[16,32] (WMMA f32) ---------
__global__ __launch_bounds__(256) void gat_gemm2(const float* __restrict__ h,
                                                 const float* __restrict__ W2,
                                                 float* __restrict__ h2, int N) {
  __shared__ float sW[16 * 32];
  lds_load_matrix(W2, sW, 16 * 32, /*dim0=*/32, /*dim1=*/16, /*stride0=*/32);

  const int wave = threadIdx.x >> 5;
  const int lane = threadIdx.x & 31;
  const int half = lane >> 4;
  const int n = lane & 15;
  const int rowbase = (blockIdx.x * 8 + wave) * 16;

  int rowa = rowbase + n;
  if (rowa >= N) rowa = N - 1;
  const float* hrow = h + (size_t)rowa * 16 + 2 * half;

  v8f c0 = {}, c1 = {};
#pragma unroll
  for (int k0 = 0; k0 < 16; k0 += 4) {
    float2 t = *(const float2*)(hrow + k0);
    v2f a;
    a.x = t.x;
    a.y = t.y;
    v2f b0, b1;
    b0.x = sW[(k0 + 2 * half + 0) * 32 + n];
    b0.y = sW[(k0 + 2 * half + 1) * 32 + n];
    b1.x = sW[(k0 + 2 * half + 0) * 32 + 16 + n];
    b1.y = sW[(k0 + 2 * half + 1) * 32 + 16 + n];
    c0 = wmma_f32_4(a, b0, c0);
    c1 = wmma_f32_4(a, b1, c1);
  }
  float* o = h2 + (size_t)(rowbase + 8 * half) * 32 + n;
  if (rowbase + 16 <= N) {  // fast path: full tile
#pragma unroll
    for (int r = 0; r < 8; ++r) {
      o[(size_t)r * 32] = c0[r];
      o[(size_t)r * 32 + 16] = c1[r];
    }
  } else {
#pragma unroll
    for (int r = 0; r < 8; ++r) {
      int rowo = rowbase + r + 8 * half;
      if (rowo < N) {
        h2[(size_t)rowo * 32 + n] = c0[r];
        h2[(size_t)rowo * 32 + 16 + n] = c1[r];
      }
    }
  }
}

// ---------------- per-node attention scores a_s = h.att_src, a_d = h.att_dst ---
template <int F>
__global__ void gat_scores(const float* __restrict__ h,
                           const float* __restrict__ asv,
                           const float* __restrict__ adv, float* __restrict__ as_,
                           float* __restrict__ ad_, int N) {
  int i = blockIdx.x * blockDim.x + threadIdx.x;
  if (i >= N) return;
  float s = 0.f, d = 0.f;
#pragma unroll
  for (int j = 0; j < F; ++j) {
    float v = h[(size_t)i * F + j];
    s += v * asv[j];
    d += v * adv[j];
  }
  as_[i] = s;
  ad_[i] = d;
}

// ---------------- edge pass 1: segment max of leaky_relu scores ----------------
__global__ void gat_edge_max(const int* __restrict__ ei, int E, int N,
                             const float* __restrict__ as_,
                             const float* __restrict__ ad_,
                             float* __restrict__ m) {
  int i = blockIdx.x * blockDim.x + threadIdx.x;
  if (i >= E + N) return;
  __builtin_prefetch(ei + i + 16384, 0, 0);
  int s, d;
  if (i < E) {
    s = ei[i];
    d = ei[E + i];
  } else {
    s = d = i - E;  // self loop
  }
  atomicMaxF(m + d, lrelu(as_[s] + ad_[d]));
}

// ---------------- edge pass 2: segment sum of exp(e - max) ---------------------
__global__ void gat_edge_expsum(const int* __restrict__ ei, int E, int N,
                                const float* __restrict__ as_,
                                const float* __restrict__ ad_,
                                const float* __restrict__ m,
                                float* __restrict__ den) {
  int i = blockIdx.x * blockDim.x + threadIdx.x;
  if (i >= E + N) return;
  __builtin_prefetch(ei + i + 16384, 0, 0);
  int s, d;
  if (i < E) {
    s = ei[i];
    d = ei[E + i];
  } else {
    s = d = i - E;
  }
  float ex = __expf(lrelu(as_[s] + ad_[d]) - m[d]);
  atomicAdd(den + d, ex);
}

// ---------------- edge pass 3: out[dst] += alpha * h[src] ----------------------
template <int F>
__global__ void gat_edge_agg(const int* __restrict__ ei, int E, int N,
                             const float* __restrict__ as_,
                             const float* __restrict__ ad_,
                             const float* __restrict__ m,
                             const float* __restrict__ den,
                             const float* __restrict__ h,
                             float* __restrict__ agg) {
  int i = blockIdx.x * blockDim.x + threadIdx.x;
  if (i >= E + N) return;
  __builtin_prefetch(ei + i + 16384, 0, 0);
  int s, d;
  if (i < E) {
    s = ei[i];
    d = ei[E + i];
  } else {
    s = d = i - E;
  }
  float alpha = __expf(lrelu(as_[s] + ad_[d]) - m[d]) / den[d];
  const float4* hs = (const float4*)(h + (size_t)s * F);
  float* o = agg + (size_t)d * F;
#pragma unroll
  for (int f = 0; f < F / 4; ++f) {
    float4 v = hs[f];
    atomicAdd(o + 4 * f + 0, v.x * alpha);
    atomicAdd(o + 4 * f + 1, v.y * alpha);
    atomicAdd(o + 4 * f + 2, v.z * alpha);
    atomicAdd(o + 4 * f + 3, v.w * alpha);
  }
}

// ---------------- layer-1 epilogue: + bias, ReLU (in place) --------------------
__global__ void gat_bias_relu16(float* __restrict__ a, const float* __restrict__ b,
                                int N) {
  int i = blockIdx.x * blockDim.x + threadIdx.x;
  if (i >= 16 * N) return;
  float v = a[i] + b[i & 15];
  a[i] = v > 0.f ? v : 0.f;
}

// ---------------- final: + bias, log_softmax over 32 classes -------------------
__global__ void gat_logsoftmax32(const float* __restrict__ agg,
                                 const float* __restrict__ b,
                                 float* __restrict__ out, int N) {
  int i = blockIdx.x * blockDim.x + threadIdx.x;
  if (i >= N) return;
  float v[32];
  float mx = -INFINITY;
#pragma unroll
  for (int j = 0; j < 32; ++j) {
    v[j] = agg[(size_t)i * 32 + j] + b[j];
    mx = fmaxf(mx, v[j]);
  }
  float s = 0.f;
#pragma unroll
  for (int j = 0; j < 32; ++j) s += __expf(v[j] - mx);
  float ls = __logf(s);
#pragma unroll
  for (int j = 0; j < 32; ++j) out[(size_t)i * 32 + j] = v[j] - mx - ls;
}

// ---------------- host launcher ------------------------------------------------
extern "C" void kernel_launch(void* const* d_in, const int* in_sizes, int n_in,
                              void* d_out, int out_size, void* d_ws,
                              size_t ws_size, hipStream_t stream) {
  const float* x = (const float*)d_in[0];
  const int* ei = (const int*)d_in[1];  // [2,E] flattened: src row then dst row
  const float* W1 = (const float*)d_in[2];
  const float* att_s1 = (const float*)d_in[3];
  const float* att_d1 = (const float*)d_in[4];
  const float* b1 = (const float*)d_in[5];
  const float* W2 = (const float*)d_in[6];
  const float* att_s2 = (const float*)d_in[7];
  const float* att_d2 = (const float*)d_in[8];
  const float* b2 = (const float*)d_in[9];

  const int N = in_sizes[0] / 512;
  const int E = in_sizes[1] / 2;
  const int E2 = E + N;

  // workspace layout (floats)
  float* ws = (float*)d_ws;
  float* h1 = ws;             ws += (size_t)N * 16;
  float* as1 = ws;            ws += N;
  float* ad1 = ws;            ws += N;
  float* m1 = ws;             ws += N;
  float* den1 = ws;           ws += N;
  float* agg1 = ws;           ws += (size_t)N * 16;  // becomes hrelu in place
  float* h2 = ws;             ws += (size_t)N * 32;
  float* as2 = ws;            ws += N;
  float* ad2 = ws;            ws += N;
  float* m2 = ws;             ws += N;
  float* den2 = ws;           ws += N;
  float* agg2 = ws;           ws += (size_t)N * 32;

  float* out = (float*)d_out;

  const int T = 256;
  const int gN = (N + T - 1) / T;
  const int gE = (E2 + T - 1) / T;
  const int gG = (N + 127) / 128;  // 8 waves x 16 rows per block
  const int g32N = (32 * N + T - 1) / T;
  const int g16N = (16 * N + T - 1) / T;

  gat_init<<<g32N, T, 0, stream>>>(m1, den1, m2, den2, agg1, agg2, N);

  // layer 1
  gat_gemm1<<<gG, T, 0, stream>>>(x, W1, h1, N);
  gat_scores<16><<<gN, T, 0, stream>>>(h1, att_s1, att_d1, as1, ad1, N);
  gat_edge_max<<<gE, T, 0, stream>>>(ei, E, N, as1, ad1, m1);
  gat_edge_expsum<<<gE, T, 0, stream>>>(ei, E, N, as1, ad1, m1, den1);
  gat_edge_agg<16><<<gE, T, 0, stream>>>(ei, E, N, as1, ad1, m1, den1, h1, agg1);
  gat_bias_relu16<<<g16N, T, 0, stream>>>(agg1, b1, N);

  // layer 2
  gat_gemm2<<<gG, T, 0, stream>>>(agg1, W2, h2, N);
  gat_scores<32><<<gN, T, 0, stream>>>(h2, att_s2, att_d2, as2, ad2, N);
  gat_edge_max<<<gE, T, 0, stream>>>(ei, E, N, as2, ad2, m2);
  gat_edge_expsum<<<gE, T, 0, stream>>>(ei, E, N, as2, ad2, m2, den2);
  gat_edge_agg<32><<<gE, T, 0, stream>>>(ei, E, N, as2, ad2, m2, den2, h2, agg2);

  gat_logsoftmax32<<<gN, T, 0, stream>>>(agg2, b2, out, N);
}